// FFN_Experts_71820443124299
// MI455X (gfx1250) — compile-verified
//
#include <hip/hip_runtime.h>
#include <math.h>

#define B_  16
#define S_  4096
#define E_  16
#define NE_ 64
#define F_  256
#define TOKENS (B_ * S_)

typedef __attribute__((ext_vector_type(2))) float v2f;
typedef __attribute__((ext_vector_type(8))) float v8f;

// LDS layout (floats), rows padded to avoid bank conflicts
#define XS_STRIDE  258              // 256 + 2
#define W1_STRIDE  258              // w1t[h][f], 64 rows
#define W2_STRIDE  66               // w2t[f][h], 256 rows
#define HS_STRIDE  66               // hs[wave][m][h], 16 rows per wave
#define XS_FLOATS  (128 * XS_STRIDE)
#define W1_FLOATS  (64 * W1_STRIDE)
#define W2_FLOATS  (256 * W2_STRIDE)
#define HS_FLOATS  (8 * 16 * HS_STRIDE)
#define LDS_FLOATS (XS_FLOATS + W1_FLOATS + W2_FLOATS + HS_FLOATS)

__device__ __forceinline__ float gelu_f(float t) {
    const float c = 0.7978845608028654f;  // sqrt(2/pi)
    return 0.5f * t * (1.0f + tanhf(c * (t + 0.044715f * t * t * t)));
}

// ---- Kernel 1: colsum[b][f] = sum_s x[b][s][f] ------------------------------
__global__ void colsum_kernel(const float* __restrict__ x, float* __restrict__ colsum) {
    const int b = blockIdx.x;       // 16 blocks
    const int f = threadIdx.x;      // 256 threads
    const float* xp = x + (size_t)b * S_ * F_ + f;
    float acc = 0.0f;
    for (int s = 0; s < S_; ++s) acc += xp[(size_t)s * F_];
    colsum[b * F_ + f] = acc;
}

// ---- Kernel 2: gate[b][e] = softmax_e(colsum[b]@Wg/S + bg) ------------------
__global__ void gate_kernel(const float* __restrict__ colsum,
                            const float* __restrict__ Wg,
                            const float* __restrict__ bg,
                            float* __restrict__ gate) {
    __shared__ float sc[B_][E_];
    const int tid = threadIdx.x;    // 256 threads: (b,e) pairs
    const int b = tid >> 4;
    const int e = tid & 15;
    float acc = 0.0f;
    for (int f = 0; f < F_; ++f) acc += colsum[b * F_ + f] * Wg[f * E_ + e];
    const float sco = acc / (float)S_ + bg[e];
    sc[b][e] = sco;
    __syncthreads();
    float m = -1e30f;
    for (int j = 0; j < E_; ++j) m = fmaxf(m, sc[b][j]);
    float den = 0.0f;
    for (int j = 0; j < E_; ++j) den += __expf(sc[b][j] - m);
    gate[b * E_ + e] = __expf(sco - m) / den;
}

// ---- Kernel 3: fused all-expert FFN with gate-weighted combine --------------
// block = 256 threads = 8 waves; each wave owns a 16-token tile (128 tok/block)
__global__ __launch_bounds__(256, 1)
void moe_ffn_kernel(const float* __restrict__ x,
                    const float* __restrict__ W1,
                    const float* __restrict__ b1,
                    const float* __restrict__ W2,
                    const float* __restrict__ b2,
                    const float* __restrict__ gate,
                    float* __restrict__ out) {
    extern __shared__ float lds[];
    float* xs  = lds;                     // [128][XS_STRIDE]
    float* w1t = xs  + XS_FLOATS;         // [64][W1_STRIDE]   (h-major: w1t[h][f])
    float* w2t = w1t + W1_FLOATS;         // [256][W2_STRIDE]  (f-major: w2t[f][h])
    float* hs  = w2t + W2_FLOATS;         // [8][16][HS_STRIDE]

    const int tid      = threadIdx.x;
    const int wave     = tid >> 5;
    const int lane     = tid & 31;
    const int laneMod  = lane & 15;
    const int laneHalf = lane >> 4;

    const int tokBlock = blockIdx.x * 128;          // S_ % 128 == 0 -> no batch straddle
    const int batch    = tokBlock / S_;

    // ---- gate weights for this batch, loaded once ----
    float gvals[E_];
#pragma unroll
    for (int e = 0; e < E_; ++e) gvals[e] = gate[batch * E_ + e];

    // ---- stage x tile once (coalesced) ----
    {
        const float* xg = x + (size_t)tokBlock * F_;
        for (int i = tid; i < 128 * F_; i += 256) {
            const int r = i >> 8;
            const int c = i & 255;
            xs[r * XS_STRIDE + c] = xg[i];
        }
    }

    // prefetch expert 0 weights into WGP cache while we queue up
    {
        const float* nw1 = W1;
        const float* nw2 = W2;
#pragma unroll
        for (int i = 0; i < 2; ++i) {
            __builtin_prefetch(nw1 + (tid + i * 256) * 32, 0, 3);   // 128B granules
            __builtin_prefetch(nw2 + (tid + i * 256) * 32, 0, 3);
        }
    }

    // persistent gate-weighted output accumulators: 16 N-tiles of 16
    v8f outacc[16] = {};

    for (int e = 0; e < E_; ++e) {
        __syncthreads();   // previous expert's compute done before overwrite
        // ---- stage W1[e] transposed -> w1t[h][f], W2[e] transposed -> w2t[f][h]
        {
            const float* w1g = W1 + (size_t)e * F_ * NE_;   // [256 f][64 h]
            for (int i = tid; i < F_ * NE_; i += 256) {
                const int f = i >> 6;
                const int h = i & 63;
                w1t[h * W1_STRIDE + f] = w1g[i];
            }
            const float* w2g = W2 + (size_t)e * NE_ * F_;   // [64 h][256 f]
            for (int i = tid; i < NE_ * F_; i += 256) {
                const int h = i >> 8;
                const int f = i & 255;
                w2t[f * W2_STRIDE + h] = w2g[i];
            }
        }
        __syncthreads();

        // ---- prefetch NEXT expert's weights into WGP$ (overlaps with compute) ----
        if (e + 1 < E_) {
            const float* nw1 = W1 + (size_t)(e + 1) * F_ * NE_;
            const float* nw2 = W2 + (size_t)(e + 1) * NE_ * F_;
#pragma unroll
            for (int i = 0; i < 2; ++i) {
                __builtin_prefetch(nw1 + (tid + i * 256) * 32, 0, 3);
                __builtin_prefetch(nw2 + (tid + i * 256) * 32, 0, 3);
            }
        }

        // ---- GEMM1: h16x64 = x16x256 @ W1e  (f32 WMMA 16x16x4, K outer) ----
        v8f acc1[4] = {};
        const float* arow = xs + (wave * 16 + laneMod) * XS_STRIDE + laneHalf * 2;
        for (int k0 = 0; k0 < F_; k0 += 4) {
            const v2f a = *(const v2f*)(arow + k0);
#pragma unroll
            for (int n = 0; n < 4; ++n) {
                const v2f bf = *(const v2f*)(w1t + (n * 16 + laneMod) * W1_STRIDE +
                                             k0 + laneHalf * 2);
                acc1[n] = __builtin_amdgcn_wmma_f32_16x16x4_f32(
                    false, a, false, bf, (short)0, acc1[n], false, false);
            }
        }

        // ---- bias + GELU, spill h to per-wave LDS tile ----
        float* hw = hs + wave * (16 * HS_STRIDE);
#pragma unroll
        for (int n = 0; n < 4; ++n) {
            const float b1v = b1[e * NE_ + n * 16 + laneMod];
#pragma unroll
            for (int v = 0; v < 8; ++v) {
                const float t = acc1[n][v] + b1v;
                hw[(laneHalf * 8 + v) * HS_STRIDE + n * 16 + laneMod] = gelu_f(t);
            }
        }

        // ---- GEMM2: o16x256 = h16x64 @ W2e; gate-weighted accumulate ----
        const float ge = gvals[e];
        const float* hrow = hw + laneMod * HS_STRIDE + laneHalf * 2;
        for (int n = 0; n < 16; ++n) {
            v8f acc2 = {};
#pragma unroll
            for (int k0 = 0; k0 < NE_; k0 += 4) {
                const v2f a  = *(const v2f*)(hrow + k0);
                const v2f bf = *(const v2f*)(w2t + (n * 16 + laneMod) * W2_STRIDE +
                                             k0 + laneHalf * 2);
                acc2 = __builtin_amdgcn_wmma_f32_16x16x4_f32(
                    false, a, false, bf, (short)0, acc2, false, false);
            }
#pragma unroll
            for (int v = 0; v < 8; ++v) outacc[n][v] += ge * acc2[v];
        }
    }

    // ---- fold gate-weighted b2 bias and write out ----
#pragma unroll
    for (int n = 0; n < 16; ++n) {
        const int f = n * 16 + laneMod;
        float bias = 0.0f;
#pragma unroll
        for (int e = 0; e < E_; ++e) bias += gvals[e] * b2[e * F_ + f];
#pragma unroll
        for (int v = 0; v < 8; ++v) {
            const int m = laneHalf * 8 + v;
            out[(size_t)(tokBlock + wave * 16 + m) * F_ + f] = outacc[n][v] + bias;
        }
    }
}

extern "C" void kernel_launch(void* const* d_in, const int* in_sizes, int n_in,
                              void* d_out, int out_size, void* d_ws, size_t ws_size,
                              hipStream_t stream) {
    const float* x  = (const float*)d_in[0];
    const float* W1 = (const float*)d_in[1];
    const float* b1 = (const float*)d_in[2];
    const float* W2 = (const float*)d_in[3];
    const float* b2 = (const float*)d_in[4];
    const float* Wg = (const float*)d_in[5];
    const float* bg = (const float*)d_in[6];
    float* out = (float*)d_out;

    float* colsum = (float*)d_ws;             // B_*F_ floats
    float* gate   = colsum + B_ * F_;         // B_*E_ floats

    colsum_kernel<<<B_, F_, 0, stream>>>(x, colsum);
    gate_kernel<<<1, 256, 0, stream>>>(colsum, Wg, bg, gate);

    const size_t ldsBytes = (size_t)LDS_FLOATS * sizeof(float);  // ~293 KB of 320 KB/WGP
    moe_ffn_kernel<<<TOKENS / 128, 256, ldsBytes, stream>>>(x, W1, b1, W2, b2, gate, out);
}